// LinearGraph_27951647163110
// MI455X (gfx1250) — compile-verified
//
#include <hip/hip_runtime.h>
#include <hip/hip_bf16.h>

typedef float v2f __attribute__((ext_vector_type(2)));
typedef float v8f __attribute__((ext_vector_type(8)));

#define HID 128

// ---------------- degree / normalization ----------------

__global__ void deg_init_kernel(float* __restrict__ deg, int n) {
    int i = blockIdx.x * blockDim.x + threadIdx.x;
    if (i < n) deg[i] = 1.0f;  // self-loop contributes 1 to every node's degree
}

__global__ void deg_edges_kernel(const long long* __restrict__ ei, float* __restrict__ deg, int E) {
    int e = blockIdx.x * blockDim.x + threadIdx.x;
    if (e < E) {
        int d = (int)ei[(size_t)E + e];  // dst row of edge_index
        atomicAdd(&deg[d], 1.0f);
    }
}

__global__ void make_dinv_kernel(float* __restrict__ deg, int n) {
    int i = blockIdx.x * blockDim.x + threadIdx.x;
    if (i < n) {
        float d = deg[i];
        deg[i] = d > 0.0f ? __frsqrt_rn(d) : 0.0f;  // in place: deg -> dinv
    }
}

// ---------------- fp32 WMMA GEMM: out[M x NCOL] = A[M x 128] @ B[128 x NCOL] (+ bias) ----------------
// B is staged once per 256-thread block into LDS in a (k-pair, n) float2 layout:
//   Bl[k/2][n] = (B[k][n], B[k+1][n])  with a 16-float2 (128B) row skew so the
//   hi=0 half-wave (banks 0..31) and hi=1 half-wave (banks 32..63) never conflict.
// Fragment layouts per ISA 7.12.2 for V_WMMA_F32_16X16X4_F32:
//   A: lane<16 -> row m0+lane, {A[r][k+0],A[r][k+1]}; lane>=16 -> {A[r][k+2],A[r][k+3]}
//      => one float2 global load at col = k + 2*hi.
//   B: lane fragment = Bl[k/2 + hi][n0+lo]  => one ds_load_b64.
//   C/D: VGPR r -> row m0 + r + 8*hi, col n0 + lo.

template <int NT, bool BIAS>
__global__ __launch_bounds__(256) void gemm_wmma_kernel(
    const float* __restrict__ A, const float* __restrict__ B,
    const float* __restrict__ bias, float* __restrict__ out, int M) {
    constexpr int KD = 128;
    constexpr int NCOL = NT * 16;
    constexpr int STR2 = NCOL + 16;  // float2 row stride with 128-byte bank skew
    __shared__ float2 Bl[(KD / 2) * STR2];

    // ---- stage B into LDS (all 8 waves share it) ----
    for (int idx = threadIdx.x; idx < (KD / 2) * NCOL; idx += 256) {
        const int kp = idx / NCOL;
        const int n = idx - kp * NCOL;
        float2 v;
        v.x = B[(size_t)(2 * kp) * NCOL + n];
        v.y = B[(size_t)(2 * kp + 1) * NCOL + n];
        Bl[kp * STR2 + n] = v;
    }
    __syncthreads();

    const int wave = threadIdx.x >> 5;
    const int lane = threadIdx.x & 31;
    const int tile = blockIdx.x * 8 + wave;
    const int tiles = (M + 15) >> 4;
    if (tile >= tiles) return;  // wave-uniform: EXEC stays all-1 inside the WMMA region

    const int m0 = tile << 4;
    const int hi = lane >> 4;
    const int lo = lane & 15;

    v8f acc[NT];
    const v8f zero = {0.f, 0.f, 0.f, 0.f, 0.f, 0.f, 0.f, 0.f};
#pragma unroll
    for (int t = 0; t < NT; ++t) acc[t] = zero;

    const float* arow = A + (size_t)(m0 + lo) * KD + 2 * hi;  // float2-aligned
    const float2* bbase = Bl + (size_t)hi * STR2 + lo;

#pragma unroll 4
    for (int k = 0; k < KD; k += 4) {
        const float2 af = *(const float2*)(arow + k);
        v2f a;
        a.x = af.x;
        a.y = af.y;
        const float2* brow = bbase + (k >> 1) * STR2;
#pragma unroll
        for (int t = 0; t < NT; ++t) {
            const float2 bf = brow[t * 16];
            v2f b;
            b.x = bf.x;
            b.y = bf.y;
            acc[t] = __builtin_amdgcn_wmma_f32_16x16x4_f32(
                /*neg_a=*/false, a, /*neg_b=*/false, b,
                /*c_mod=*/(short)0, acc[t], /*reuse_a=*/false, /*reuse_b=*/false);
        }
    }

#pragma unroll
    for (int t = 0; t < NT; ++t) {
        const int col = t * 16 + lo;
        const float bv = BIAS ? bias[col] : 0.0f;
#pragma unroll
        for (int r = 0; r < 8; ++r) {
            const int row = m0 + r + 8 * hi;
            out[(size_t)row * NCOL + col] = acc[t][r] + bv;
        }
    }
}

// ---------------- aggregation ----------------

// agg[i][:] = bias[:] + dinv[i]^2 * t[i][:]   (self-loop term fused with bias)
__global__ void agg_init_kernel(const float* __restrict__ t, const float* __restrict__ dinv,
                                const float* __restrict__ bias, float* __restrict__ agg, int n) {
    int idx = blockIdx.x * blockDim.x + threadIdx.x;  // n * (HID/4) threads
    if (idx >= n * (HID / 4)) return;
    const int i = idx >> 5;          // node
    const int j4 = (idx & 31) << 2;  // float4 column
    const float di = dinv[i];
    const float d2 = di * di;
    const float4 tv = *(const float4*)(t + (size_t)i * HID + j4);
    const float4 bv = *(const float4*)(bias + j4);
    float4 o;
    o.x = bv.x + d2 * tv.x;
    o.y = bv.y + d2 * tv.y;
    o.z = bv.z + d2 * tv.z;
    o.w = bv.w + d2 * tv.w;
    *(float4*)(agg + (size_t)i * HID + j4) = o;
}

// One wave per edge: agg[dst][:] += dinv[src]*dinv[dst] * t[src][:]
__global__ void scatter_edges_kernel(const long long* __restrict__ ei, const float* __restrict__ dinv,
                                     const float* __restrict__ t, float* __restrict__ agg, int E) {
    const int e = blockIdx.x * (blockDim.x >> 5) + (threadIdx.x >> 5);
    if (e >= E) return;
    const int lane = threadIdx.x & 31;
    const int s = (int)ei[e];
    const int d = (int)ei[(size_t)E + e];
    const float nrm = dinv[s] * dinv[d];
    const float4 v = *(const float4*)(t + (size_t)s * HID + lane * 4);
    float* op = agg + (size_t)d * HID + lane * 4;
    atomicAdd(op + 0, nrm * v.x);
    atomicAdd(op + 1, nrm * v.y);
    atomicAdd(op + 2, nrm * v.z);
    atomicAdd(op + 3, nrm * v.w);
}

__global__ void relu_ip_kernel(float* __restrict__ a, long long n4) {
    long long i = (long long)blockIdx.x * blockDim.x + threadIdx.x;
    if (i >= n4) return;
    float4 v = ((float4*)a)[i];
    v.x = fmaxf(v.x, 0.f);
    v.y = fmaxf(v.y, 0.f);
    v.z = fmaxf(v.z, 0.f);
    v.w = fmaxf(v.w, 0.f);
    ((float4*)a)[i] = v;
}

// ---------------- launch ----------------

extern "C" void kernel_launch(void* const* d_in, const int* in_sizes, int n_in,
                              void* d_out, int out_size, void* d_ws, size_t ws_size,
                              hipStream_t stream) {
    const float* x = (const float*)d_in[0];
    const long long* ei = (const long long*)d_in[1];
    const float* We = (const float*)d_in[2];
    const float* be = (const float*)d_in[3];
    const float* W1 = (const float*)d_in[4];
    const float* b1 = (const float*)d_in[5];
    const float* W2 = (const float*)d_in[6];
    const float* b2 = (const float*)d_in[7];
    const float* Wo = (const float*)d_in[8];
    const float* bo = (const float*)d_in[9];
    float* out = (float*)d_out;

    const int N = in_sizes[0] / HID;  // 50000
    const int E = in_sizes[1] / 2;    // 1600000

    char* ws = (char*)d_ws;
    size_t off = ((size_t)N * sizeof(float) + 255) & ~(size_t)255;
    float* dinv = (float*)ws;              // N floats (deg -> dinv in place)
    float* buf0 = (float*)(ws + off);      // N x HID
    float* buf1 = buf0 + (size_t)N * HID;  // N x HID

    const int tiles = (N + 15) / 16;
    const dim3 gemmGrid((tiles + 7) / 8);
    const int nThr = (N + 255) / 256;
    const int eThr = (E + 255) / 256;
    const int aggThr = (N * (HID / 4) + 255) / 256;
    const int edgeWaves = (E + 7) / 8;  // 8 waves (edges) per 256-thread block
    const long long n4 = (long long)N * (HID / 4);
    const int reluThr = (int)((n4 + 255) / 256);

    // degree / symmetric normalization
    deg_init_kernel<<<nThr, 256, 0, stream>>>(dinv, N);
    deg_edges_kernel<<<eThr, 256, 0, stream>>>(ei, dinv, E);
    make_dinv_kernel<<<nThr, 256, 0, stream>>>(dinv, N);

    // encoder: h = x @ We + be -> buf0
    gemm_wmma_kernel<8, true><<<gemmGrid, 256, 0, stream>>>(x, We, be, buf0, N);

    // conv1: t = h @ W1 -> buf1; agg = b1 + dinv^2*t + scatter; relu -> buf0
    gemm_wmma_kernel<8, false><<<gemmGrid, 256, 0, stream>>>(buf0, W1, nullptr, buf1, N);
    agg_init_kernel<<<aggThr, 256, 0, stream>>>(buf1, dinv, b1, buf0, N);
    scatter_edges_kernel<<<edgeWaves, 256, 0, stream>>>(ei, dinv, buf1, buf0, E);
    relu_ip_kernel<<<reluThr, 256, 0, stream>>>(buf0, n4);

    // conv2
    gemm_wmma_kernel<8, false><<<gemmGrid, 256, 0, stream>>>(buf0, W2, nullptr, buf1, N);
    agg_init_kernel<<<aggThr, 256, 0, stream>>>(buf1, dinv, b2, buf0, N);
    scatter_edges_kernel<<<edgeWaves, 256, 0, stream>>>(ei, dinv, buf1, buf0, E);
    relu_ip_kernel<<<reluThr, 256, 0, stream>>>(buf0, n4);

    // decoder: out = h @ Wo + bo  (64 classes -> 4 column tiles)
    gemm_wmma_kernel<4, true><<<gemmGrid, 256, 0, stream>>>(buf0, Wo, bo, out, N);
}